// YOLOLoss_22497038696638
// MI455X (gfx1250) — compile-verified
//
#include <hip/hip_runtime.h>
#include <hip/hip_bf16.h>
#include <math.h>

// Problem constants (match reference setup_inputs)
#define BB 32
#define AA 3
#define HH 80
#define WW 80
#define NN 50
#define HW (HH * WW)              // 6400
#define CPB 25                    // 6400 / 256 cells-chunks per (b,a)
#define NBLK (BB * AA * CPB)      // 2400 blocks in main kernel
#define PI_F 3.1415926f

typedef __attribute__((ext_vector_type(2))) float v2f;
typedef __attribute__((ext_vector_type(8))) float v8f;

__device__ __forceinline__ float sigmoidf_(float x) {
    return 1.0f / (1.0f + __expf(-x));
}

// ---------------------------------------------------------------------------
// Kernel A: per (b,n) GT box -> target cell (anchor argmax, gi, gj) + cached
// box info {cx,cy,w,h, x0,y0,x1,y1}. 1600 threads total.
// ---------------------------------------------------------------------------
__global__ void yolo_targets(const float* __restrict__ boxes,
                             const float* __restrict__ anchors,
                             float* __restrict__ tinfo,   // [B*N*8]
                             int*   __restrict__ tcell)   // [B*N]
{
    int t = blockIdx.x * blockDim.x + threadIdx.x;
    if (t >= BB * NN) return;
    const float* bx = boxes + (size_t)t * 4;
    float cx = bx[0], cy = bx[1], w = bx[2], h = bx[3];
    int gi = (int)floorf(cx * (float)WW);
    int gj = (int)floorf(cy * (float)HH);
    // anchor argmax of IoU(wh, anchor) ; first max wins (strict >)
    float best = -1.0f; int anc = 0;
    for (int a = 0; a < AA; ++a) {
        float aw = anchors[a * 2 + 0], ah = anchors[a * 2 + 1];
        float inter = fminf(w, aw) * fminf(h, ah);
        float uni   = w * h + aw * ah - inter;
        float r = inter / uni;
        if (r > best) { best = r; anc = a; }
    }
    float* ti = tinfo + (size_t)t * 8;
    ti[0] = cx; ti[1] = cy; ti[2] = w; ti[3] = h;
    ti[4] = cx - 0.5f * w; ti[5] = cy - 0.5f * h;
    ti[6] = cx + 0.5f * w; ti[7] = cy + 0.5f * h;
    tcell[t] = (anc * HH + gj) * WW + gi;
}

// ---------------------------------------------------------------------------
// Kernel B: one thread per cell. Decode + max-IoU-vs-50-GT (division free) +
// BCE + CIoU, then wave shfl reduction and a WMMA(ones-A) column-sum combine
// of the 8 wave partials. Writes 8 floats/block (slots 6,7 are zeros).
// ---------------------------------------------------------------------------
__global__ __launch_bounds__(256) void yolo_cells(
    const float* __restrict__ head,
    const float* __restrict__ anchors,
    const float* __restrict__ tinfo,
    const int*   __restrict__ tcell,
    float* __restrict__ partials)    // [NBLK*8]
{
    __shared__ float sBox[NN * 8];
    __shared__ int   sCell[NN];
    __shared__ float sWave[8][8];

    const int blk = blockIdx.x;
    const int b   = blk / (AA * CPB);
    const int r   = blk % (AA * CPB);
    const int a   = r / CPB;
    const int tid = threadIdx.x;
    const int p   = (r % CPB) * 256 + tid;   // 0..6399 within (b,a) plane
    const int gj  = p / WW, gi = p % WW;

    if (tid < NN) {
        #pragma unroll
        for (int k = 0; k < 8; ++k)
            sBox[tid * 8 + k] = tinfo[((size_t)b * NN + tid) * 8 + k];
        sCell[tid] = tcell[b * NN + tid];
    }
    const float aw = anchors[a * 2 + 0], ah = anchors[a * 2 + 1];

    const float* base = head + ((size_t)b * (AA * 6) + a * 6) * HW + p;
    // CDNA5 prefetch of the later channel planes (global_prefetch_b8)
    __builtin_prefetch(base + 3 * HW, 0, 0);
    __builtin_prefetch(base + 5 * HW, 0, 0);
    float t0 = base[0];
    float t1 = base[1 * HW];
    float t2 = base[2 * HW];
    float t3 = base[3 * HW];
    float t4 = base[4 * HW];
    float t5 = base[5 * HW];
    __syncthreads();

    // ---- decode ----
    float obj = sigmoidf_(t0);
    float cx  = 1.2f * sigmoidf_(t1) - 0.1f;
    float cy  = 1.2f * sigmoidf_(t2) - 0.1f;
    float bx  = (cx + (float)gi) / (float)WW;
    float by  = (cy + (float)gj) / (float)HH;
    float bw  = __expf(t3) * aw;
    float bh  = __expf(t4) * ah;
    float cls = sigmoidf_(t5);
    float px0 = bx - 0.5f * bw, py0 = by - 0.5f * bh;
    float px1 = bx + 0.5f * bw, py1 = by + 0.5f * bh;
    float pArea = bw * bh;
    const int myCell = (a * HH + gj) * WW + gi;

    // ---- max-IoU > 0.5 test + target membership (last match wins) ----
    bool over = false;
    int  tm = -1;
    #pragma unroll 5
    for (int n = 0; n < NN; ++n) {
        const float* e = &sBox[n * 8];
        float iw = fmaxf(fminf(px1, e[6]) - fmaxf(px0, e[4]), 0.0f);
        float ih = fmaxf(fminf(py1, e[7]) - fmaxf(py0, e[5]), 0.0f);
        float inter = iw * ih;
        float uni   = pArea + e[2] * e[3] - inter;
        over = over || (inter > 0.5f * uni);     // iou > IGNORE, no division
        if (sCell[n] == myCell) tm = n;
    }
    float tobj   = (tm >= 0) ? 1.0f : 0.0f;
    float noobjv = over ? 0.0f : 1.0f;

    // ---- BCE terms ----
    float bce1  = fminf(-__logf(obj), 100.0f);        // bce(obj, 1)
    float bce0  = fminf(-log1pf(-obj), 100.0f);       // bce(obj, 0)
    float bce_t = (tobj > 0.0f) ? bce1 : bce0;        // bce(obj, t_obj)
    float bcec  = fminf(-__logf(cls), 100.0f);        // bce(cls, 1)  (C==1, label==0)

    // ---- CIoU vs matched target box (garbage if tm<0; masked by tobj) ----
    const float* e = &sBox[(tm >= 0 ? tm : 0) * 8];
    float qcx = e[0], qcy = e[1], qw = e[2], qh = e[3];
    float iw = fmaxf(fminf(px1, e[6]) - fmaxf(px0, e[4]), 0.0f);
    float ih = fmaxf(fminf(py1, e[7]) - fmaxf(py0, e[5]), 0.0f);
    float inter = iw * ih;
    float uni = fmaxf(bw * bh + qw * qh - inter, 1e-6f);
    float iou = inter / uni;
    float dd = fabsf(bx - qcx) + fabsf(by - qcy);
    float cw = fmaxf(px1, e[6]) - fminf(px0, e[4]);
    float ch = fmaxf(py1, e[7]) - fminf(py0, e[5]);
    float dis = dd / fmaxf(cw + ch, 1e-6f);
    float a1 = atanf(bw / fmaxf(bh, 1e-6f));
    float a2 = atanf(qw / fmaxf(qh, 1e-6f));
    float v  = (4.0f / (PI_F * PI_F)) * fabsf(a1 - a2);
    float alpha = v / fmaxf(1.0f - iou + v, 1e-6f);
    float ciou  = 1.0f - iou + dis + alpha * v;

    // ---- per-thread contributions ----
    float s0 = tobj * ciou;          // sum t_obj * ciou
    float s1 = tobj * bce1;          // obj_loss
    float s2 = noobjv * bce_t;       // noobj_loss
    float s3 = tobj * bcec;          // cls numerator
    float s4 = tobj;                 // n_pos
    float s5 = tobj + noobjv;        // denom

    // ---- wave32 butterfly reduction (deterministic order) ----
    #pragma unroll
    for (int m = 16; m >= 1; m >>= 1) {
        s0 += __shfl_xor(s0, m, 32);
        s1 += __shfl_xor(s1, m, 32);
        s2 += __shfl_xor(s2, m, 32);
        s3 += __shfl_xor(s3, m, 32);
        s4 += __shfl_xor(s4, m, 32);
        s5 += __shfl_xor(s5, m, 32);
    }
    const int wave = tid >> 5, lane = tid & 31;
    if (lane == 0) {
        sWave[wave][0] = s0; sWave[wave][1] = s1; sWave[wave][2] = s2;
        sWave[wave][3] = s3; sWave[wave][4] = s4; sWave[wave][5] = s5;
        sWave[wave][6] = 0.0f; sWave[wave][7] = 0.0f;
    }
    __syncthreads();

    // ---- wave 0: combine 8 wave partials with v_wmma_f32_16x16x4_f32 ----
    // A = all-ones 16x4  =>  D[m][n] = sum over the 4 B slots of column n.
    // Column n = lane%16; lanes {q, q+16} x {v0,v1} feed column q.
    if (wave == 0) {
        int q = lane & 15, h = lane >> 4;
        v2f onesA = {1.0f, 1.0f};
        v2f bv, bv2;
        bv.x  = (q < 8) ? sWave[2 * h + 0][q] : 0.0f;   // waves 0..3
        bv.y  = (q < 8) ? sWave[2 * h + 1][q] : 0.0f;
        bv2.x = (q < 8) ? sWave[4 + 2 * h + 0][q] : 0.0f; // waves 4..7
        bv2.y = (q < 8) ? sWave[4 + 2 * h + 1][q] : 0.0f;
        v8f c = {};
        v8f d = __builtin_amdgcn_wmma_f32_16x16x4_f32(
            false, onesA, false, bv, (short)0, c, false, false);
        d = __builtin_amdgcn_wmma_f32_16x16x4_f32(
            false, onesA, false, bv2, (short)0, d, false, false);
        // every row of D equals the column totals; lane L's d[0] = total(L%16)
        if (lane < 8) partials[(size_t)blk * 8 + lane] = d[0];
    }
}

// ---------------------------------------------------------------------------
// Kernel C: deterministic final combine of 2400 block partials via one
// WMMA column-sum (one wave), then emit the two scalar losses.
// ---------------------------------------------------------------------------
__global__ void yolo_final(const float* __restrict__ partials,
                           float* __restrict__ out)
{
    __shared__ float tot[8];
    int lane = threadIdx.x;          // 0..31, single wave
    int q = lane & 15, h = lane >> 4;
    float acc = 0.0f;
    if (q < 8) {
        for (int i = h; i < NBLK; i += 2)   // fixed order per lane
            acc += partials[(size_t)i * 8 + q];
    }
    v2f onesA = {1.0f, 1.0f};
    v2f bv; bv.x = acc; bv.y = 0.0f;
    v8f c = {};
    v8f d = __builtin_amdgcn_wmma_f32_16x16x4_f32(
        false, onesA, false, bv, (short)0, c, false, false);
    if (lane < 8) tot[lane] = d[0];   // column q total = lanes {q,q+16} halves
    __syncthreads();
    if (lane == 0) {
        float ciou_s = tot[0], obj_s = tot[1], noobj_s = tot[2];
        float cls_s = tot[3], npos = tot[4], denom = tot[5];
        float box_loss = 0.05f * ciou_s / npos;
        float grid_loss = (1.5f * obj_s + 0.5f * noobj_s) / denom
                        + 0.5f * (cls_s / (npos * 1.0f));
        out[0] = box_loss;
        out[1] = grid_loss;
    }
}

// ---------------------------------------------------------------------------
extern "C" void kernel_launch(void* const* d_in, const int* in_sizes, int n_in,
                              void* d_out, int out_size, void* d_ws, size_t ws_size,
                              hipStream_t stream)
{
    const float* head    = (const float*)d_in[0];  // (32, 18, 80, 80) f32
    const float* boxes   = (const float*)d_in[1];  // (32, 50, 4) f32
    const float* anchors = (const float*)d_in[2];  // (3, 2) f32
    // d_in[3] (labels) unused: C == 1 so every label is 0.

    float* ws       = (float*)d_ws;
    float* tinfo    = ws;                                   // 32*50*8 floats
    int*   tcell    = (int*)(ws + BB * NN * 8);             // 32*50 ints
    float* partials = ws + BB * NN * 8 + BB * NN;           // 2400*8 floats

    yolo_targets<<<(BB * NN + 255) / 256, 256, 0, stream>>>(
        boxes, anchors, tinfo, tcell);
    yolo_cells<<<NBLK, 256, 0, stream>>>(
        head, anchors, tinfo, tcell, partials);
    yolo_final<<<1, 32, 0, stream>>>(partials, (float*)d_out);
}